// DotAttentionLayer_2499670966473
// MI455X (gfx1250) — compile-verified
//
#include <hip/hip_runtime.h>
#include <math.h>

typedef float v2f __attribute__((ext_vector_type(2)));
typedef float v8f __attribute__((ext_vector_type(8)));

#define FDIM 128
#define NHEADS 4
// per-head dim = 32, scale = 1/sqrt(32)
#define DOT_SCALE 0.17677669529663687f

__device__ __forceinline__ float lrelu(float x) { return x >= 0.0f ? x : 0.2f * x; }

// ---------------------------------------------------------------------------
// out[rows x 128] = act(X[rows x 128] @ W[128 x 128] + b)
// One wave computes one 16x16 output tile via V_WMMA_F32_16X16X4_F32.
// Block = 256 threads = 8 waves -> 16 rows x 128 cols per block.
// A layout (16x4 f32): lanes 0-15 hold M=lane, VGPR{0,1}=K{0,1}; lanes 16-31 K{2,3}.
// B layout (4x16 f32): mirrored; C/D: VGPR r -> row r (lanes 0-15) / r+8 (lanes 16-31).
// ---------------------------------------------------------------------------
template <bool ACT>
__global__ __launch_bounds__(256) void gemm128_wmma_kernel(
    const float* __restrict__ X, const float* __restrict__ W,
    const float* __restrict__ bias, float* __restrict__ out, int rows)
{
    const int lane = threadIdx.x & 31;
    const int wave = threadIdx.x >> 5;   // column tile 0..7
    const int half = lane >> 4;          // 0 or 1
    const int l16  = lane & 15;

    int rb = blockIdx.x * 16;
    if (rb + 16 > rows) rb = rows - 16;  // duplicate tail tile; writes identical values
    const int cb = wave * 16;

    const float* __restrict__ Arow = X + (size_t)(rb + l16) * FDIM;

    v8f c = {0.f, 0.f, 0.f, 0.f, 0.f, 0.f, 0.f, 0.f};

#pragma unroll
    for (int k0 = 0; k0 < FDIM; k0 += 4) {
        const int ka = k0 + 2 * half;
        v2f a, b;
        a.x = Arow[ka];
        a.y = Arow[ka + 1];
        b.x = W[(size_t)ka * FDIM + cb + l16];
        b.y = W[(size_t)(ka + 1) * FDIM + cb + l16];
        c = __builtin_amdgcn_wmma_f32_16x16x4_f32(
                /*neg_a=*/false, a, /*neg_b=*/false, b,
                /*c_mod=*/(short)0, c, /*reuse_a=*/false, /*reuse_b=*/false);
    }

    const float bv = bias[cb + l16];
#pragma unroll
    for (int r = 0; r < 8; ++r) {
        float val = c[r] + bv;
        if (ACT) val = lrelu(val);
        out[(size_t)(rb + r + 8 * half) * FDIM + cb + l16] = val;
    }
}

// ---------------------------------------------------------------------------
// init: mx = -inf, den = 0, res = 0
// ---------------------------------------------------------------------------
__global__ void init_kernel(float* __restrict__ mx, float* __restrict__ den,
                            float* __restrict__ res, int mh, int resn)
{
    const int i = blockIdx.x * blockDim.x + threadIdx.x;
    if (i < mh) {
        mx[i]  = __int_as_float(0xff800000); // -inf
        den[i] = 0.0f;
    }
    if (i < resn) res[i] = 0.0f;
}

// sign-split float atomic max against -inf-initialized buffer
__device__ __forceinline__ void atomic_max_f32(float* addr, float v)
{
    if (v >= 0.0f) atomicMax((int*)addr, __float_as_int(v));
    else           atomicMin((unsigned int*)addr, __float_as_uint(v));
}

// ---------------------------------------------------------------------------
// One wave per edge: dot(q[row], k[col]) per head (4 heads x 32 dims),
// then atomic segment-max into mx keyed by edge_cols.
// ---------------------------------------------------------------------------
__global__ __launch_bounds__(256) void edge_dot_kernel(
    const float* __restrict__ q, const float* __restrict__ k,
    const int* __restrict__ erow, const int* __restrict__ ecol,
    float* __restrict__ dot, float* __restrict__ mx, int E)
{
    const int lane = threadIdx.x & 31;
    const int e = blockIdx.x * 8 + (threadIdx.x >> 5);
    if (e >= E) return;

    const int r = erow[e];
    const int c = ecol[e];

    const float4 qv = *(const float4*)(q + (size_t)r * FDIM + lane * 4);
    const float4 kv = *(const float4*)(k + (size_t)c * FDIM + lane * 4);
    float p = qv.x * kv.x + qv.y * kv.y + qv.z * kv.z + qv.w * kv.w;

    // reduce within each 8-lane group (one head = 8 lanes x 4 dims)
    p += __shfl_xor(p, 1, 32);
    p += __shfl_xor(p, 2, 32);
    p += __shfl_xor(p, 4, 32);

    if ((lane & 7) == 0) {
        const int h = lane >> 3;
        const float d = p * DOT_SCALE;
        dot[(size_t)e * NHEADS + h] = d;
        atomic_max_f32(mx + (size_t)c * NHEADS + h, d);
    }
}

// ---------------------------------------------------------------------------
// ex = exp(dot - mx[col]); in-place into dot; atomic segment-sum into den.
// One thread per (edge, head).
// ---------------------------------------------------------------------------
__global__ void exp_den_kernel(const int* __restrict__ ecol,
                               const float* __restrict__ mx,
                               float* __restrict__ dot,
                               float* __restrict__ den, int EH)
{
    const int i = blockIdx.x * blockDim.x + threadIdx.x;
    if (i >= EH) return;
    const int e = i >> 2;
    const int h = i & 3;
    const int c = ecol[e];
    const float ex = __expf(dot[i] - mx[(size_t)c * NHEADS + h]);
    dot[i] = ex;
    atomicAdd(den + (size_t)c * NHEADS + h, ex);
}

// ---------------------------------------------------------------------------
// One wave per edge: alpha = ex/den[col]; res[row] += alpha * v[col].
// L2-resident f32 atomics (res is 25.6 MB << 192 MB L2).
// ---------------------------------------------------------------------------
__global__ __launch_bounds__(256) void aggregate_kernel(
    const float* __restrict__ v, const int* __restrict__ erow,
    const int* __restrict__ ecol, const float* __restrict__ ex,
    const float* __restrict__ den, float* __restrict__ res, int E)
{
    const int lane = threadIdx.x & 31;
    const int e = blockIdx.x * 8 + (threadIdx.x >> 5);
    if (e >= E) return;

    const int r = erow[e];
    const int c = ecol[e];
    const int h = lane >> 3;

    const float alpha = ex[(size_t)e * NHEADS + h] / den[(size_t)c * NHEADS + h];
    const float4 vv = *(const float4*)(v + (size_t)c * FDIM + lane * 4);
    float* o = res + (size_t)r * FDIM + lane * 4;
    atomicAdd(o + 0, alpha * vv.x);
    atomicAdd(o + 1, alpha * vv.y);
    atomicAdd(o + 2, alpha * vv.z);
    atomicAdd(o + 3, alpha * vv.w);
}

// ---------------------------------------------------------------------------
extern "C" void kernel_launch(void* const* d_in, const int* in_sizes, int n_in,
                              void* d_out, int out_size, void* d_ws, size_t ws_size,
                              hipStream_t stream)
{
    (void)n_in; (void)out_size; (void)ws_size;

    const float* query  = (const float*)d_in[0];
    const float* memory = (const float*)d_in[1];
    const int*   erow   = (const int*)d_in[2];
    const int*   ecol   = (const int*)d_in[3];
    const float* Wq = (const float*)d_in[4];
    const float* bq = (const float*)d_in[5];
    const float* Wk = (const float*)d_in[6];
    const float* bk = (const float*)d_in[7];
    const float* Wv = (const float*)d_in[8];
    const float* bv = (const float*)d_in[9];
    const float* Wo = (const float*)d_in[10];
    const float* bo = (const float*)d_in[11];
    float* out = (float*)d_out;

    const int N = in_sizes[0] / FDIM;
    const int M = in_sizes[1] / FDIM;
    const int E = in_sizes[2];

    // workspace layout (floats)
    float* ws  = (float*)d_ws;
    float* q   = ws;  ws += (size_t)N * FDIM;
    float* kb  = ws;  ws += (size_t)M * FDIM;
    float* vb  = ws;  ws += (size_t)M * FDIM;
    float* dot = ws;  ws += (size_t)E * NHEADS;   // becomes ex in-place
    float* mx  = ws;  ws += (size_t)M * NHEADS;
    float* den = ws;  ws += (size_t)M * NHEADS;
    float* res = ws;  ws += (size_t)N * FDIM;

    const int mh   = M * NHEADS;
    const int resn = N * FDIM;
    const int initn = (mh > resn) ? mh : resn;

    init_kernel<<<(initn + 255) / 256, 256, 0, stream>>>(mx, den, res, mh, resn);

    gemm128_wmma_kernel<true ><<<(N + 15) / 16, 256, 0, stream>>>(query,  Wq, bq, q,  N);
    gemm128_wmma_kernel<true ><<<(M + 15) / 16, 256, 0, stream>>>(memory, Wk, bk, kb, M);
    gemm128_wmma_kernel<true ><<<(M + 15) / 16, 256, 0, stream>>>(memory, Wv, bv, vb, M);

    edge_dot_kernel<<<(E + 7) / 8, 256, 0, stream>>>(q, kb, erow, ecol, dot, mx, E);
    exp_den_kernel<<<(E * NHEADS + 255) / 256, 256, 0, stream>>>(ecol, mx, dot, den, E * NHEADS);
    aggregate_kernel<<<(E + 7) / 8, 256, 0, stream>>>(vb, erow, ecol, dot, den, res, E);

    gemm128_wmma_kernel<false><<<(N + 15) / 16, 256, 0, stream>>>(res, Wo, bo, out, N);
}